// MyBertAttnBPWordPieceCRF_20555713478674
// MI455X (gfx1250) — compile-verified
//
#include <hip/hip_runtime.h>
#include <math.h>

typedef float v2f __attribute__((ext_vector_type(2)));
typedef float v8f __attribute__((ext_vector_type(8)));

#define B_   128
#define S_   512
#define D_   768
#define T_   51
#define NI_  26
#define NS_  122
#define ROWS_   (B_ * T_)     /* 6528 */
#define MTILES_ (ROWS_ / 16)  /* 408  */
#define LPAD 772              /* padded LDS row stride (floats): 772%64==4 -> no bank conflicts */

// ---------------------------------------------------------------------------
// Intent head: res[128,768] @ W_id[768,26] + b_id  (tiny, scalar kernel)
// ---------------------------------------------------------------------------
__global__ __launch_bounds__(256) void intent_kernel(
    const float* __restrict__ res, const float* __restrict__ W_id,
    const float* __restrict__ b_id, float* __restrict__ out)
{
    int idx = blockIdx.x * blockDim.x + threadIdx.x;
    if (idx >= B_ * NI_) return;
    int b = idx / NI_;
    int j = idx - b * NI_;
    const float* r = res + (size_t)b * D_;
    float acc = 0.f;
    for (int d = 0; d < D_; ++d)
        acc = fmaf(r[d], W_id[(size_t)d * NI_ + j], acc);
    out[idx] = acc + b_id[j];
}

// ---------------------------------------------------------------------------
// Fused: ragged gather -> Q = g0 @ W_attn (WMMA f32 16x16x4) -> bilinear
// scores -> softmax(2) / len==1 select -> word = pool + res -> word @ W_slot
// One workgroup (8 waves) per 16-row tile of the (B*T) row dimension.
// ---------------------------------------------------------------------------
__global__ __launch_bounds__(256) void word_slot_kernel(
    const float* __restrict__ res_all, const float* __restrict__ res,
    const int*  __restrict__ tstart,   const int*  __restrict__ slen,
    const float* __restrict__ W_attn,  const float* __restrict__ W_slot,
    const float* __restrict__ b_slot,  float* __restrict__ out_sf)
{
    __shared__ float g0s[16][LPAD];   // gathered g0 rows; later overwritten with `word`
    __shared__ float g1s[16][LPAD];   // gathered g1 rows
    __shared__ float sc0[16], sc1[16], wg0[16], wg1[16];
    __shared__ int   rowB[16], rowS0[16], rowS1[16], rowLen1[16];

    const int tid  = threadIdx.x;
    const int lane = tid & 31;
    const int wave = tid >> 5;          // 0..7
    const int half = (lane >> 4) & 1;   // 0: lanes 0-15, 1: lanes 16-31
    const int l16  = lane & 15;
    const int kA   = 2 * half;          // K sub-offset for this half (A/B frag layout)
    const int mblk = blockIdx.x;

    // ---- Phase 0a: per-row metadata for this 16-row tile ----
    if (tid < 16) {
        int R  = mblk * 16 + tid;
        int bb = R / T_;
        int tt = R - bb * T_;
        int st = tstart[bb * T_ + tt];
        int ln = slen[bb * T_ + tt];
        rowB[tid]    = bb;
        rowS0[tid]   = st;
        rowS1[tid]   = st + ln;          // start+len < 507 < 512, always in range
        rowLen1[tid] = (ln == 1);
        sc0[tid] = 0.f;
        sc1[tid] = 0.f;
    }
    __syncthreads();

    // ---- Phase 0b: ragged gather of g0/g1 rows into LDS (float4) ----
    for (int idx = tid; idx < 16 * (D_ / 4); idx += 256) {
        int m = idx / (D_ / 4);
        int q = idx - m * (D_ / 4);
        const float4* s0p = (const float4*)(res_all + (size_t)(rowB[m] * S_ + rowS0[m]) * D_);
        const float4* s1p = (const float4*)(res_all + (size_t)(rowB[m] * S_ + rowS1[m]) * D_);
        ((float4*)&g0s[m][0])[q] = s0p[q];
        ((float4*)&g1s[m][0])[q] = s1p[q];
    }
    __syncthreads();

    // ---- Phase 1: Q = g0_tile(16x768) @ W_attn(768x768), fused with
    //      bilinear score accumulation (Q never materialized). ----
    float s0r[8], s1r[8];
#pragma unroll
    for (int i = 0; i < 8; ++i) { s0r[i] = 0.f; s1r[i] = 0.f; }

    for (int nt = 0; nt < 6; ++nt) {
        const int nb = wave * 96 + nt * 16;
        const int c  = nb + l16;                // this lane's N column
        v8f acc = {};
        for (int k0 = 0; k0 < D_; k0 += 4) {
            v2f a, b;
            a.x = g0s[l16][k0 + kA];            // A: M=l16, K=k0+kA(+1)
            a.y = g0s[l16][k0 + kA + 1];
            b.x = W_attn[(size_t)(k0 + kA) * D_ + c];       // B: K rows, N=c
            b.y = W_attn[(size_t)(k0 + kA + 1) * D_ + c];
            acc = __builtin_amdgcn_wmma_f32_16x16x4_f32(
                false, a, false, b, (short)0, acc, false, false);
        }
        // C layout: vgpr i -> M = i + 8*half, N = c. Dot against g0/g1 now.
#pragma unroll
        for (int i = 0; i < 8; ++i) {
            int   m = i + 8 * half;
            float q = acc[i];
            s0r[i] = fmaf(q, g0s[m][c], s0r[i]);
            s1r[i] = fmaf(q, g1s[m][c], s1r[i]);
        }
    }

    // Reduce partial scores across the 16 lanes of each half, then merge waves.
#pragma unroll
    for (int i = 0; i < 8; ++i) {
        float a0 = s0r[i], a1 = s1r[i];
        for (int off = 8; off >= 1; off >>= 1) {
            a0 += __shfl_xor(a0, off, 32);
            a1 += __shfl_xor(a1, off, 32);
        }
        if (l16 == 0) {
            int m = i + 8 * half;
            atomicAdd(&sc0[m], a0);
            atomicAdd(&sc1[m], a1);
        }
    }
    __syncthreads();

    // ---- Phase 2: scaled 2-way softmax, len==1 hard-select ----
    if (tid < 16) {
        const float scale = 0.03608439182435161f;   // 768^-0.5
        float a  = sc0[tid] * scale;
        float b  = sc1[tid] * scale;
        float mx = fmaxf(a, b);
        float e0 = __expf(a - mx);
        float e1 = __expf(b - mx);
        float iv = 1.f / (e0 + e1);
        float w0 = e0 * iv, w1 = e1 * iv;
        if (rowLen1[tid]) { w0 = 1.f; w1 = 0.f; }
        wg0[tid] = w0;
        wg1[tid] = w1;
    }
    __syncthreads();

    // ---- Phase 2b: word = w0*g0 + w1*g1 + res[b]  (overwrite g0s) ----
    for (int idx = tid; idx < 16 * D_; idx += 256) {
        int m = idx / D_;
        int d = idx - m * D_;
        float w = wg0[m] * g0s[m][d] + wg1[m] * g1s[m][d]
                + res[(size_t)rowB[m] * D_ + d];
        g0s[m][d] = w;
    }
    __syncthreads();

    // ---- Phase 3: res_sf = word(16x768) @ W_slot(768x122) + b_slot ----
    {
        const int  nb = wave * 16;
        const int  c  = nb + l16;
        const bool cv = (c < NS_);
        const int  cc = cv ? c : (NS_ - 1);
        const float msk = cv ? 1.f : 0.f;
        v8f acc = {};
        for (int k0 = 0; k0 < D_; k0 += 4) {
            v2f a, b;
            a.x = g0s[l16][k0 + kA];
            a.y = g0s[l16][k0 + kA + 1];
            b.x = W_slot[(size_t)(k0 + kA) * NS_ + cc] * msk;
            b.y = W_slot[(size_t)(k0 + kA + 1) * NS_ + cc] * msk;
            acc = __builtin_amdgcn_wmma_f32_16x16x4_f32(
                false, a, false, b, (short)0, acc, false, false);
        }
        if (cv) {
            float bias = b_slot[c];
#pragma unroll
            for (int i = 0; i < 8; ++i) {
                int m = i + 8 * half;
                int R = mblk * 16 + m;
                out_sf[(size_t)R * NS_ + c] = acc[i] + bias;
            }
        }
    }
}

// ---------------------------------------------------------------------------
extern "C" void kernel_launch(void* const* d_in, const int* in_sizes, int n_in,
                              void* d_out, int out_size, void* d_ws, size_t ws_size,
                              hipStream_t stream)
{
    (void)in_sizes; (void)n_in; (void)out_size; (void)d_ws; (void)ws_size;

    const float* res_all = (const float*)d_in[0];
    const float* res     = (const float*)d_in[1];
    const int*   tstart  = (const int*)  d_in[2];
    const int*   slen    = (const int*)  d_in[3];
    const float* W_attn  = (const float*)d_in[4];
    const float* W_id    = (const float*)d_in[5];
    const float* b_id    = (const float*)d_in[6];
    const float* W_slot  = (const float*)d_in[7];
    const float* b_slot  = (const float*)d_in[8];

    float* out_id = (float*)d_out;            // [128, 26]
    float* out_sf = out_id + B_ * NI_;        // [128, 51, 122]

    intent_kernel<<<(B_ * NI_ + 255) / 256, 256, 0, stream>>>(res, W_id, b_id, out_id);
    word_slot_kernel<<<MTILES_, 256, 0, stream>>>(
        res_all, res, tstart, slen, W_attn, W_slot, b_slot, out_sf);
}